// GINEncoder_9861244912035
// MI455X (gfx1250) — compile-verified
//
#include <hip/hip_runtime.h>

typedef __attribute__((ext_vector_type(2))) float v2f;
typedef __attribute__((ext_vector_type(8))) float v8f;

#define DIN   128
#define HWID  256      // hidden width H
#define DOUTC 128
#define BN_EPS 1e-5f

__device__ __forceinline__ void atomicAddF(float* p, float v) {
  __hip_atomic_fetch_add(p, v, __ATOMIC_RELAXED, __HIP_MEMORY_SCOPE_AGENT);
}

// ---------------------------------------------------------------------------
// Scatter-add: agg[dst[e]*D + c .. +3] += feat[src[e]*D + c .. +3]
// One thread handles one float4 chunk of one edge. Accumulators are L2-
// resident (51/102 MB < 192 MB L2), so the atomic RMW stays on-chip.
// ---------------------------------------------------------------------------
__global__ void scatter_add_kernel(const float* __restrict__ feat,
                                   const int* __restrict__ src,
                                   const int* __restrict__ dst,
                                   float* __restrict__ agg,
                                   int E, int D) {
  long long tid = (long long)blockIdx.x * blockDim.x + threadIdx.x;
  int chunks = D >> 2;
  long long total = (long long)E * chunks;
  if (tid >= total) return;
  int e = (int)(tid / chunks);
  int c = ((int)(tid % chunks)) << 2;
  const float4 v = *(const float4*)(feat + (long long)src[e] * D + c);
  float* o = agg + (long long)dst[e] * D + c;
  atomicAddF(o + 0, v.x);
  atomicAddF(o + 1, v.y);
  atomicAddF(o + 2, v.z);
  atomicAddF(o + 3, v.w);
}

// ---------------------------------------------------------------------------
// fp32 WMMA GEMM:  C[M,DOUT] = opA(A[M,K]) @ W[K,DOUT] + bias, optional ReLU.
// opA (when FUSE) = relu(a * ascale[k] + ashift[k])   (fused BN+ReLU)
//
// One wave32 computes a 16x32 output slab = TWO adjacent 16x16 C tiles with
// V_WMMA_F32_16X16X4_F32, sharing one A fragment per K-step (2x WMMA density
// per A load). K and DOUT are template params -> all addressing is immediate
// offsets; unroll-by-4 batches 4 A-loads + 16 B-loads ahead of 8 WMMAs.
//
// A fragment (16x4, ISA layout): lanes 0-15 hold K=k+0,k+1 for row M=lane;
// lanes 16-31 hold K=k+2,k+3 for row M=lane-16. B mirrors (4x16 by column).
// C/D layout: lanes 0-15 -> rows 0..7 (vgpr i = row i), lanes 16-31 -> 8..15.
// Requires M % 16 == 0 (N = 100000 = 6250*16). EXEC all-ones per live wave.
// ---------------------------------------------------------------------------
template <int K, int DOUT, bool FUSE, bool RELU>
__global__ void __launch_bounds__(128)
wmma_gemm_f32_kernel(const float* __restrict__ A,
                     const float* __restrict__ W,
                     const float* __restrict__ bias,
                     const float* __restrict__ ascale,
                     const float* __restrict__ ashift,
                     float* __restrict__ C, int M) {
  const int wave = threadIdx.x >> 5;
  const int lane = threadIdx.x & 31;
  constexpr int NPAIR = DOUT / 32;              // 16x32 slabs per row-block
  const int pair  = blockIdx.x * 4 + wave;
  const int total = (M >> 4) * NPAIR;
  if (pair >= total) return;                    // uniform per-wave branch

  const int tm   = pair / NPAIR;
  const int tp   = pair % NPAIR;
  const int half = lane >> 4;                   // 0: lanes 0-15, 1: 16-31
  const int l15  = lane & 15;
  const int row  = tm * 16 + l15;               // A row owned by this lane
  const int col0 = tp * 32 + l15;               // first C column
  const int col1 = col0 + 16;                   // second C column
  const int kOff = half * 2;

  const float* __restrict__ Arow = A + (long long)row * K + kOff;
  const float* __restrict__ Wp   = W + (long long)kOff * DOUT;
  v8f acc0 = {};
  v8f acc1 = {};

#pragma unroll 4
  for (int k = 0; k < K; k += 4) {
    v2f a = *(const v2f*)(Arow + k);            // 8B aligned (k+kOff even)
    if (FUSE) {
      const int k0 = k + kOff;
      v2f sc = *(const v2f*)(ascale + k0);
      v2f sh = *(const v2f*)(ashift + k0);
      a.x = fmaxf(a.x * sc.x + sh.x, 0.0f);
      a.y = fmaxf(a.y * sc.y + sh.y, 0.0f);
    }
    const float* wk = Wp + (long long)k * DOUT;
    v2f b0, b1;
    b0.x = wk[col0];
    b0.y = wk[DOUT + col0];
    b1.x = wk[col1];
    b1.y = wk[DOUT + col1];
    acc0 = __builtin_amdgcn_wmma_f32_16x16x4_f32(
        false, a, false, b0, (short)0, acc0, false, false);
    acc1 = __builtin_amdgcn_wmma_f32_16x16x4_f32(
        false, a, false, b1, (short)0, acc1, false, false);
  }

  const float bv0 = bias[col0];
  const float bv1 = bias[col1];
  float* __restrict__ Crow = C + (long long)(tm * 16 + half * 8) * DOUT;
#pragma unroll
  for (int i = 0; i < 8; ++i) {
    float v0 = acc0[i] + bv0;
    float v1 = acc1[i] + bv1;
    if (RELU) {
      v0 = fmaxf(v0, 0.0f);
      v1 = fmaxf(v1, 0.0f);
    }
    Crow[(long long)i * DOUT + col0] = v0;
    Crow[(long long)i * DOUT + col1] = v1;
  }
}

// ---------------------------------------------------------------------------
// BatchNorm statistics: per-column sum / sum-of-squares. blockDim.x == D.
// Each block privately accumulates a row stripe, then one atomic per column.
// ---------------------------------------------------------------------------
__global__ void bn_stats_kernel(const float* __restrict__ t, int M, int D,
                                float* __restrict__ sums,
                                float* __restrict__ sumsq,
                                int rowsPerBlock) {
  const int col = threadIdx.x;
  const int r0  = blockIdx.x * rowsPerBlock;
  float s = 0.0f, sq = 0.0f;
  for (int i = 0; i < rowsPerBlock; ++i) {
    const int r = r0 + i;
    if (r < M) {
      const float v = t[(long long)r * D + col];
      s  += v;
      sq += v * v;
    }
  }
  atomicAddF(&sums[col], s);
  atomicAddF(&sumsq[col], sq);
}

__global__ void bn_finalize_kernel(const float* __restrict__ sums,
                                   const float* __restrict__ sumsq,
                                   const float* __restrict__ gamma,
                                   const float* __restrict__ beta,
                                   float* __restrict__ scale,
                                   float* __restrict__ shift,
                                   int D, float invN) {
  const int c = blockIdx.x * blockDim.x + threadIdx.x;
  if (c >= D) return;
  const float mu  = sums[c] * invN;
  const float var = sumsq[c] * invN - mu * mu;      // biased variance
  const float sc  = gamma[c] * rsqrtf(var + BN_EPS);
  scale[c] = sc;
  shift[c] = beta[c] - mu * sc;
}

__global__ void zero_kernel(float* __restrict__ p, int n) {
  const int i = blockIdx.x * blockDim.x + threadIdx.x;
  if (i < n) p[i] = 0.0f;
}

// ---------------------------------------------------------------------------
extern "C" void kernel_launch(void* const* d_in, const int* in_sizes, int n_in,
                              void* d_out, int out_size, void* d_ws, size_t ws_size,
                              hipStream_t stream) {
  const float* x     = (const float*)d_in[0];
  const int*   src   = (const int*)  d_in[1];
  const int*   dst   = (const int*)  d_in[2];
  const float* l0_w1 = (const float*)d_in[3];
  const float* l0_b1 = (const float*)d_in[4];
  const float* l0_g  = (const float*)d_in[5];
  const float* l0_be = (const float*)d_in[6];
  const float* l0_w2 = (const float*)d_in[7];
  const float* l0_b2 = (const float*)d_in[8];
  const float* l1_w1 = (const float*)d_in[9];
  const float* l1_b1 = (const float*)d_in[10];
  const float* l1_g  = (const float*)d_in[11];
  const float* l1_be = (const float*)d_in[12];
  const float* l1_w2 = (const float*)d_in[13];
  const float* l1_b2 = (const float*)d_in[14];

  const int N = in_sizes[0] / DIN;
  const int E = in_sizes[1];
  float* out = (float*)d_out;

  // Workspace layout: buf1 [N*H], buf2 [N*H], stats [4*H]
  float* buf1  = (float*)d_ws;
  float* buf2  = buf1 + (size_t)N * HWID;
  float* stats = buf2 + (size_t)N * HWID;
  float* sums  = stats;
  float* sumsq = stats + HWID;
  float* scale = stats + 2 * HWID;
  float* shift = stats + 3 * HWID;

  const int rpb = 256;                       // BN rows per block
  const int bnBlocks = (N + rpb - 1) / rpb;

  // ======================= Layer 0 =======================
  // agg0 = x  (h = x + segment_sum(x[src], dst)); accumulate into d_out region
  hipMemcpyAsync(out, x, (size_t)N * DIN * sizeof(float),
                 hipMemcpyDeviceToDevice, stream);
  {
    long long tot = (long long)E * (DIN / 4);
    scatter_add_kernel<<<(unsigned)((tot + 255) / 256), 256, 0, stream>>>(
        x, src, dst, out, E, DIN);
  }
  // t0 = h0 @ l0_w1 + b1   ->  buf1 [N, 256]
  {
    int pairs = (N / 16) * (HWID / 32);
    wmma_gemm_f32_kernel<DIN, HWID, false, false>
        <<<(pairs + 3) / 4, 128, 0, stream>>>(
            out, l0_w1, l0_b1, nullptr, nullptr, buf1, N);
  }
  // BN stats over nodes (256 features)
  zero_kernel<<<(2 * HWID + 255) / 256, 256, 0, stream>>>(stats, 2 * HWID);
  bn_stats_kernel<<<bnBlocks, HWID, 0, stream>>>(buf1, N, HWID, sums, sumsq, rpb);
  bn_finalize_kernel<<<1, HWID, 0, stream>>>(sums, sumsq, l0_g, l0_be,
                                             scale, shift, HWID, 1.0f / (float)N);
  // h0_out = relu( relu(BN(t0)) @ l0_w2 + b2 )  ->  buf2 [N, 256]
  {
    int pairs = (N / 16) * (HWID / 32);
    wmma_gemm_f32_kernel<HWID, HWID, true, true>
        <<<(pairs + 3) / 4, 128, 0, stream>>>(
            buf1, l0_w2, l0_b2, scale, shift, buf2, N);
  }

  // ======================= Layer 1 =======================
  // agg1 = h0_out; accumulate into buf1
  hipMemcpyAsync(buf1, buf2, (size_t)N * HWID * sizeof(float),
                 hipMemcpyDeviceToDevice, stream);
  {
    long long tot = (long long)E * (HWID / 4);
    scatter_add_kernel<<<(unsigned)((tot + 255) / 256), 256, 0, stream>>>(
        buf2, src, dst, buf1, E, HWID);
  }
  // t1 = h1 @ l1_w1 + b1  ->  buf2 reused as [N, 128]
  {
    int pairs = (N / 16) * (DOUTC / 32);
    wmma_gemm_f32_kernel<HWID, DOUTC, false, false>
        <<<(pairs + 3) / 4, 128, 0, stream>>>(
            buf1, l1_w1, l1_b1, nullptr, nullptr, buf2, N);
  }
  // BN stats (128 features)
  zero_kernel<<<(2 * HWID + 255) / 256, 256, 0, stream>>>(stats, 2 * HWID);
  bn_stats_kernel<<<bnBlocks, DOUTC, 0, stream>>>(buf2, N, DOUTC, sums, sumsq, rpb);
  bn_finalize_kernel<<<1, DOUTC, 0, stream>>>(sums, sumsq, l1_g, l1_be,
                                              scale, shift, DOUTC, 1.0f / (float)N);
  // out = relu( relu(BN(t1)) @ l1_w2 + b2 )  ->  d_out [N, 128]
  {
    int pairs = (N / 16) * (DOUTC / 32);
    wmma_gemm_f32_kernel<DIN, DOUTC, true, true>
        <<<(pairs + 3) / 4, 128, 0, stream>>>(
            buf2, l1_w2, l1_b2, scale, shift, out, N);
  }
}